// WordRNN_19267223290562
// MI455X (gfx1250) — compile-verified
//
#include <hip/hip_runtime.h>
#include <hip/hip_bf16.h>
#include <math.h>

// ---------------------------------------------------------------------------
// WordRNN: embedding gather -> LSTM(512) -> LSTM(512) -> Dense(32000)
// B=16, T=256, E=256, H=512, V=32000.
// GEMMs on V_WMMA_F32_16X16X4_F32; A-panels staged in LDS via the gfx1250
// async-to-LDS path (ASYNCcnt); persistent LSTM recurrence with a
// device-wide spin barrier.
// ---------------------------------------------------------------------------

typedef __attribute__((ext_vector_type(2))) float v2f;
typedef __attribute__((ext_vector_type(8))) float v8f;
typedef __attribute__((ext_vector_type(4))) int   v4i_t;

#define BN 16      // batch
#define TN 256     // seq len
#define EN 256     // embed dim
#define HN 512     // hidden
#define N4 2048    // 4*H
#define VN 32000   // vocab
#define MROWS (BN * TN)   // 4096 GEMM rows (b*T + t)

// ------------------- async global->LDS copy (gfx1250 path) -----------------
#if defined(__has_builtin)
#if __has_builtin(__builtin_amdgcn_global_load_async_to_lds_b128)
#define ASYNC_LDS 1
#endif
#endif

typedef __attribute__((address_space(1))) v4i_t* gptr_v4i;
typedef __attribute__((address_space(3))) v4i_t* lptr_v4i;

__device__ __forceinline__ void copy16_to_lds(float* lds_dst, const float* gsrc) {
#ifdef ASYNC_LDS
    __builtin_amdgcn_global_load_async_to_lds_b128(
        (gptr_v4i)(gsrc),
        (lptr_v4i)(lds_dst),
        /*offset=*/0, /*cpol=*/0);
#else
    *(float4*)lds_dst = *(const float4*)gsrc;
#endif
}

__device__ __forceinline__ void wait_lds_copies() {
#ifdef ASYNC_LDS
    asm volatile("s_wait_asynccnt 0x0" ::: "memory");
#endif
}

// -------------------------- device-wide barrier ----------------------------
__device__ __forceinline__ void grid_barrier(int* bar, int nblk) {
    __syncthreads();
    if (threadIdx.x == 0) {
        int gen = __hip_atomic_load(&bar[1], __ATOMIC_ACQUIRE, __HIP_MEMORY_SCOPE_AGENT);
        int arrived = __hip_atomic_fetch_add(&bar[0], 1, __ATOMIC_ACQ_REL, __HIP_MEMORY_SCOPE_AGENT);
        if (arrived == nblk - 1) {
            __hip_atomic_store(&bar[0], 0, __ATOMIC_RELAXED, __HIP_MEMORY_SCOPE_AGENT);
            __hip_atomic_fetch_add(&bar[1], 1, __ATOMIC_ACQ_REL, __HIP_MEMORY_SCOPE_AGENT);
        } else {
            while (__hip_atomic_load(&bar[1], __ATOMIC_ACQUIRE, __HIP_MEMORY_SCOPE_AGENT) == gen) {
                __builtin_amdgcn_s_sleep(1);
            }
        }
    }
    __syncthreads();
}

__global__ void bar_init_kernel(int* bar) {
    bar[0] = 0;
    bar[1] = 0;
}

// --------------------------- embedding gather ------------------------------
__global__ void embed_gather_kernel(const int* __restrict__ tokens,
                                    const float* __restrict__ emb,
                                    float* __restrict__ x0) {
    int m = blockIdx.x;
    int e = threadIdx.x;
    int tok = tokens[m];
    x0[(size_t)m * EN + e] = emb[(size_t)tok * EN + e];
}

// ------------------- fp32 WMMA GEMM:  D = A*B + bias -----------------------
// A: [M,K] row-major (K <= 512).  B: [K,N] row-major.  D: [M,N] row-major.
// Block = 128 threads (4 waves). Block tile: 16 rows x 256 cols.
// Wave w owns 4 column tiles (64 cols); A panel staged in LDS, shared by all.
// grid.x = N/256, grid.y = M/16.
__global__ void gemm_wmma_f32_kernel(const float* __restrict__ A,
                                     const float* __restrict__ Bm,
                                     const float* __restrict__ bias,
                                     float* __restrict__ D,
                                     int M, int N, int K) {
    __shared__ __align__(16) float As[16 * 512];

    const int lane = threadIdx.x & 31;
    const int wave = threadIdx.x >> 5;
    const int m0   = blockIdx.y * 16;

    // stage A panel [16 x K] into LDS, coalesced 16B chunks
    for (int i = threadIdx.x * 4; i < 16 * K; i += 128 * 4) {
        int r  = i / K;
        int kc = i - r * K;
        copy16_to_lds(&As[i], &A[(size_t)(m0 + r) * K + kc]);
    }
    wait_lds_copies();
    __syncthreads();

    const int colb = blockIdx.x * 256 + wave * 64 + (lane & 15);
    const int kh   = (lane >> 4) * 2;            // lanes 16-31 carry K = 2,3
    const int rb   = (lane >> 4) * 8;            // C/D rows 8..15 on hi half

    v8f acc0 = {}, acc1 = {}, acc2 = {}, acc3 = {};
    const float* arow = As + (lane & 15) * K;

    for (int k0 = 0; k0 < K; k0 += 4) {
        v2f a = *(const v2f*)(arow + k0 + kh);   // ds_load_b64, bank-friendly
        const float* brow0 = Bm + (size_t)(k0 + kh) * N;
        const float* brow1 = Bm + (size_t)(k0 + kh + 1) * N;
        v2f b;
        b.x = brow0[colb];      b.y = brow1[colb];
        acc0 = __builtin_amdgcn_wmma_f32_16x16x4_f32(false, a, false, b, (short)0, acc0, false, false);
        b.x = brow0[colb + 16]; b.y = brow1[colb + 16];
        acc1 = __builtin_amdgcn_wmma_f32_16x16x4_f32(false, a, false, b, (short)0, acc1, false, false);
        b.x = brow0[colb + 32]; b.y = brow1[colb + 32];
        acc2 = __builtin_amdgcn_wmma_f32_16x16x4_f32(false, a, false, b, (short)0, acc2, false, false);
        b.x = brow0[colb + 48]; b.y = brow1[colb + 48];
        acc3 = __builtin_amdgcn_wmma_f32_16x16x4_f32(false, a, false, b, (short)0, acc3, false, false);
    }

    float bv0 = bias ? bias[colb]      : 0.0f;
    float bv1 = bias ? bias[colb + 16] : 0.0f;
    float bv2 = bias ? bias[colb + 32] : 0.0f;
    float bv3 = bias ? bias[colb + 48] : 0.0f;
    #pragma unroll
    for (int r = 0; r < 8; ++r) {
        float* drow = D + (size_t)(m0 + rb + r) * N;
        drow[colb]      = acc0[r] + bv0;
        drow[colb + 16] = acc1[r] + bv1;
        drow[colb + 32] = acc2[r] + bv2;
        drow[colb + 48] = acc3[r] + bv3;
    }
}

// ------------------------- persistent LSTM layer ---------------------------
// xw:   [B,T,4H] precomputed input path (row index b*T+t)
// U:    [H,4H] recurrent weights (L2-resident, 4 MB)
// hseq: [B,T,H] output sequence
// 32 blocks x 128 threads = 128 waves; wave g owns z columns [16g, 16g+16).
// The h panel [16 x 512] is staged into LDS once per block per timestep.
__global__ void lstm_recurrent_kernel(const float* __restrict__ xw,
                                      const float* __restrict__ U,
                                      float* __restrict__ hseq,
                                      float* __restrict__ hbuf,
                                      float* __restrict__ cbuf,
                                      float* __restrict__ zbuf,
                                      int* __restrict__ bar) {
    __shared__ __align__(16) float Hs[BN * HN];   // 32 KB

    const int lane  = threadIdx.x & 31;
    const int wave  = threadIdx.x >> 5;
    const int gwave = blockIdx.x * 4 + wave;            // 0..127
    const int col   = gwave * 16 + (lane & 15);         // z column (0..2047)
    const int kh    = (lane >> 4) * 2;
    const int rb    = (lane >> 4) * 8;
    const int tid   = blockIdx.x * blockDim.x + threadIdx.x;   // 0..4095
    const int nthr  = gridDim.x * blockDim.x;
    const int nblk  = gridDim.x;

    // zero initial h, c state
    for (int i = tid; i < BN * HN; i += nthr) {
        hbuf[i] = 0.0f;
        cbuf[i] = 0.0f;
    }
    grid_barrier(bar, nblk);

    for (int t = 0; t < TN; ++t) {
        // --- stage h panel into LDS (async, tracked by ASYNCcnt) ------------
        for (int i = threadIdx.x * 4; i < BN * HN; i += 128 * 4) {
            copy16_to_lds(&Hs[i], &hbuf[i]);
        }
        wait_lds_copies();
        __syncthreads();

        // --- z tile = xw[:, t, cols] + h @ U[:, cols] -----------------------
        v8f acc;
        #pragma unroll
        for (int r = 0; r < 8; ++r) {
            acc[r] = xw[(size_t)(rb + r) * (TN * N4) + (size_t)t * N4 + col];
        }
        const float* hrow = Hs + (lane & 15) * HN;
        for (int k0 = 0; k0 < HN; k0 += 4) {
            v2f a = *(const v2f*)(hrow + k0 + kh);
            v2f b;
            b.x = U[(size_t)(k0 + kh) * N4 + col];
            b.y = U[(size_t)(k0 + kh + 1) * N4 + col];
            acc = __builtin_amdgcn_wmma_f32_16x16x4_f32(
                false, a, false, b, (short)0, acc, false, false);
        }
        #pragma unroll
        for (int r = 0; r < 8; ++r) {
            zbuf[(rb + r) * N4 + col] = acc[r];
        }
        grid_barrier(bar, nblk);

        // --- fused gate update: c = f*c + i*g ; h = o*tanh(c) ---------------
        for (int i = tid; i < BN * HN; i += nthr) {
            int b = i >> 9;             // i / 512
            int j = i & (HN - 1);       // i % 512
            float zi = zbuf[b * N4 + j];
            float zf = zbuf[b * N4 + HN + j];
            float zg = zbuf[b * N4 + 2 * HN + j];
            float zo = zbuf[b * N4 + 3 * HN + j];
            float ig = 1.0f / (1.0f + __expf(-zi));
            float fg = 1.0f / (1.0f + __expf(-zf));
            float gg = tanhf(zg);
            float og = 1.0f / (1.0f + __expf(-zo));
            float c  = fg * cbuf[i] + ig * gg;
            cbuf[i] = c;
            float h  = og * tanhf(c);
            hbuf[i] = h;
            hseq[(size_t)b * TN * HN + (size_t)t * HN + j] = h;
        }
        grid_barrier(bar, nblk);
    }
}

// ------------------------------- launcher ----------------------------------
extern "C" void kernel_launch(void* const* d_in, const int* in_sizes, int n_in,
                              void* d_out, int out_size, void* d_ws, size_t ws_size,
                              hipStream_t stream) {
    const int*   tokens = (const int*)d_in[0];
    const float* emb    = (const float*)d_in[1];
    const float* W1     = (const float*)d_in[2];
    const float* U1     = (const float*)d_in[3];
    const float* b1     = (const float*)d_in[4];
    const float* W2     = (const float*)d_in[5];
    const float* U2     = (const float*)d_in[6];
    const float* b2     = (const float*)d_in[7];
    const float* Wd     = (const float*)d_in[8];
    const float* bd     = (const float*)d_in[9];
    float* out = (float*)d_out;

    // workspace carve-out (256B aligned slabs)
    char* ws = (char*)d_ws;
    size_t off = 0;
    auto carve = [&](size_t nfloats) -> float* {
        float* p = (float*)(ws + off);
        off += ((nfloats * sizeof(float) + 255) / 256) * 256;
        return p;
    };
    float* x0   = carve((size_t)MROWS * EN);     //  4 MB  embedding output
    float* xwb  = carve((size_t)MROWS * N4);     // 32 MB  precomputed input path
    float* hs1  = carve((size_t)MROWS * HN);     //  8 MB  layer-1 h sequence
    float* hs2  = carve((size_t)MROWS * HN);     //  8 MB  layer-2 h sequence
    float* hbuf = carve((size_t)BN * HN);        // current h
    float* cbuf = carve((size_t)BN * HN);        // current c
    float* zbuf = carve((size_t)BN * N4);        // current z (4 gates)
    int*   bar  = (int*)(ws + off); off += 256;  // barrier {count, generation}

    // 1) embedding gather
    embed_gather_kernel<<<MROWS, EN, 0, stream>>>(tokens, emb, x0);

    // 2) layer-1 input path: xw = x0 @ W1 + b1   (M=4096, N=2048, K=256)
    gemm_wmma_f32_kernel<<<dim3(N4 / 256, MROWS / 16), 128, 0, stream>>>(
        x0, W1, b1, xwb, MROWS, N4, EN);

    // 3) layer-1 recurrence (persistent, grid-synced)
    bar_init_kernel<<<1, 1, 0, stream>>>(bar);
    lstm_recurrent_kernel<<<32, 128, 0, stream>>>(xwb, U1, hs1, hbuf, cbuf, zbuf, bar);

    // 4) layer-2 input path: xw = hs1 @ W2 + b2  (M=4096, N=2048, K=512)
    gemm_wmma_f32_kernel<<<dim3(N4 / 256, MROWS / 16), 128, 0, stream>>>(
        hs1, W2, b2, xwb, MROWS, N4, HN);

    // 5) layer-2 recurrence
    bar_init_kernel<<<1, 1, 0, stream>>>(bar);
    lstm_recurrent_kernel<<<32, 128, 0, stream>>>(xwb, U2, hs2, hbuf, cbuf, zbuf, bar);

    // 6) dense head: logits = hs2 @ Wd + bd      (M=4096, N=32000, K=512)
    gemm_wmma_f32_kernel<<<dim3(VN / 256, MROWS / 16), 128, 0, stream>>>(
        hs2, Wd, bd, out, MROWS, VN, HN);
}